// AugmChamferLoss_66271345377749
// MI455X (gfx1250) — compile-verified
//
#include <hip/hip_runtime.h>
#include <math.h>

typedef float v2f __attribute__((ext_vector_type(2)));
typedef float v8f __attribute__((ext_vector_type(8)));

#define NPTS 4096
#define NB 8
#define JSPLIT 4                       // column-range split for occupancy
#define JCOLS (NPTS / JSPLIT)          // 1024 columns per block
#define JTILES (JCOLS / 16)            // 64 j-tiles per block
#define ROWS_PER_WAVE 32               // two 16-row A tiles per wave
#define WAVES_PER_BLOCK 8
#define ROWS_PER_BLOCK (ROWS_PER_WAVE * WAVES_PER_BLOCK)   // 256
#define NROWBLOCKS (NPTS / ROWS_PER_BLOCK)                 // 16

#define INF_BITS 0x7F800000            // +inf as int; non-negative floats are
                                       // monotone under signed-int compare

// lane-xor shuffle via ds_swizzle group-of-32 mode: and=0x1F, or=0, xor=m
#define XSHUF(x, m) __builtin_amdgcn_ds_swizzle((x), (((m) << 10) | 0x1f))

static __device__ __forceinline__ int imin2(int a, int b) { return a < b ? a : b; }

// ---------------------------------------------------------------------------
// Init workspace: row-min and column-min accumulators to +inf (bit pattern)
// ---------------------------------------------------------------------------
__global__ void chamfer_init(int* __restrict__ gMins) {
    int i = blockIdx.x * blockDim.x + threadIdx.x;
    if (i < 2 * NB * NPTS) gMins[i] = INF_BITS;
}

// ---------------------------------------------------------------------------
// Main kernel. Each wave owns 32 gts rows (two 16x4 A tiles) and sweeps 64
// pred column tiles. Per j-tile: one B96 load + one B-operand build feeds TWO
// V_WMMA_F32_16X16X4_F32 ops, each producing a full 16x16 squared-distance
// tile:   A=[x0,x1,x2,||x||^2], B=[-2y0,-2y1,-2y2,1], C=||y||^2 broadcast
//   =>    D[i][j] = ||x_i||^2 + ||y_j||^2 - 2 x.y   (>= 0)
// All min reductions run on the int bit patterns (no canonicalize ops).
// ---------------------------------------------------------------------------
__global__ __launch_bounds__(256) void chamfer_main(
    const float* __restrict__ preds,   // [B, N, 3]  (columns j)
    const float* __restrict__ gts,     // [B, N, 3]  (rows i)
    int* __restrict__ gColMin,         // [B*N]  min over i of P[b,i,j] (bits)
    int* __restrict__ gRowMin)         // [B*N]  min over j of P[b,i,j] (bits)
{
    __shared__ int ldsColMin[JCOLS];

    const int tid   = threadIdx.x;
    const int lane  = tid & 31;
    const int wave  = tid >> 5;
    const int b     = blockIdx.y;
    const int r0    = (blockIdx.x * WAVES_PER_BLOCK + wave) * ROWS_PER_WAVE;
    const int jbase = blockIdx.z * JCOLS;
    const bool lo   = lane < 16;
    const int  l16  = lane & 15;

    for (int i = tid; i < JCOLS; i += 256) ldsColMin[i] = INF_BITS;
    __syncthreads();

    // --- Load two A tiles (gts rows r0..r0+31), fold ||x||^2 into K=3 ---
    // 32-bit A 16x4 layout: lanes 0-15: VGPR0=K0, VGPR1=K1; lanes 16-31: K2, K3
    const float* xpA = gts + ((size_t)b * NPTS + r0 + l16) * 3;
    const float* xpB = xpA + 16 * 3;
    float xa0 = xpA[0], xa1 = xpA[1], xa2 = xpA[2];
    float xb0 = xpB[0], xb1 = xpB[1], xb2 = xpB[2];
    float xxa = xa0 * xa0 + xa1 * xa1 + xa2 * xa2;
    float xxb = xb0 * xb0 + xb1 * xb1 + xb2 * xb2;
    v2f aA, aB;
    aA.x = lo ? xa0 : xa2;  aA.y = lo ? xa1 : xxa;
    aB.x = lo ? xb0 : xb2;  aB.y = lo ? xb1 : xxb;

    int rowMinA[8], rowMinB[8];
#pragma unroll
    for (int v = 0; v < 8; ++v) { rowMinA[v] = INF_BITS; rowMinB[v] = INF_BITS; }

    const float* pb = preds + ((size_t)b * NPTS + jbase) * 3;

    // Software-pipelined B-tile point loads (wrapped index => branch-free)
    float y0 = pb[l16 * 3 + 0], y1 = pb[l16 * 3 + 1], y2 = pb[l16 * 3 + 2];

    for (int jt = 0; jt < JTILES; ++jt) {
        const int nj = (jt + 1) & (JTILES - 1);
        const float* np = pb + (size_t)(nj * 16 + l16) * 3;
        float ny0 = np[0], ny1 = np[1], ny2 = np[2];

        float yy = y0 * y0 + y1 * y1 + y2 * y2;
        v2f bm;
        bm.x = lo ? -2.f * y0 : -2.f * y2;  // K0 | K2
        bm.y = lo ? -2.f * y1 : 1.0f;       // K1 | K3

        v8f cA = {yy, yy, yy, yy, yy, yy, yy, yy};
        v8f cB = cA;

        v8f dA = __builtin_amdgcn_wmma_f32_16x16x4_f32(
            false, aA, false, bm, (short)0, cA, false, false);
        v8f dB = __builtin_amdgcn_wmma_f32_16x16x4_f32(
            false, aB, false, bm, (short)0, cB, false, false);

        int tmin = imin2(__builtin_bit_cast(int, dA[0]),
                         __builtin_bit_cast(int, dB[0]));
#pragma unroll
        for (int v = 0; v < 8; ++v) {
            int da = __builtin_bit_cast(int, dA[v]);
            int db = __builtin_bit_cast(int, dB[v]);
            rowMinA[v] = imin2(rowMinA[v], da);
            rowMinB[v] = imin2(rowMinB[v], db);
            tmin = imin2(tmin, imin2(da, db));
        }
        // lanes l and l+16 cover rows v / v+8 of the same column j
        atomicMin(&ldsColMin[jt * 16 + l16], tmin);

        y0 = ny0; y1 = ny1; y2 = ny2;
    }

    // --- Row-min reduce across the 16 lanes of each half (wave32) ---
    // guard-free xor shuffles via ds_swizzle (literal masks)
#pragma unroll
    for (int v = 0; v < 8; ++v) rowMinA[v] = imin2(rowMinA[v], XSHUF(rowMinA[v], 1));
#pragma unroll
    for (int v = 0; v < 8; ++v) rowMinB[v] = imin2(rowMinB[v], XSHUF(rowMinB[v], 1));
#pragma unroll
    for (int v = 0; v < 8; ++v) rowMinA[v] = imin2(rowMinA[v], XSHUF(rowMinA[v], 2));
#pragma unroll
    for (int v = 0; v < 8; ++v) rowMinB[v] = imin2(rowMinB[v], XSHUF(rowMinB[v], 2));
#pragma unroll
    for (int v = 0; v < 8; ++v) rowMinA[v] = imin2(rowMinA[v], XSHUF(rowMinA[v], 4));
#pragma unroll
    for (int v = 0; v < 8; ++v) rowMinB[v] = imin2(rowMinB[v], XSHUF(rowMinB[v], 4));
#pragma unroll
    for (int v = 0; v < 8; ++v) rowMinA[v] = imin2(rowMinA[v], XSHUF(rowMinA[v], 8));
#pragma unroll
    for (int v = 0; v < 8; ++v) rowMinB[v] = imin2(rowMinB[v], XSHUF(rowMinB[v], 8));

    // lane 0 holds rows r0+v / r0+16+v; lane 16 holds rows r0+8+v / r0+24+v
    if (l16 == 0) {
        const int base = lo ? 0 : 8;
        int* rm = gRowMin + (size_t)b * NPTS + r0 + base;
#pragma unroll
        for (int v = 0; v < 8; ++v) {
            atomicMin(&rm[v],      rowMinA[v]);
            atomicMin(&rm[16 + v], rowMinB[v]);
        }
    }

    // --- Publish this block's column mins (exact min => deterministic) ---
    __syncthreads();
    for (int i = tid; i < JCOLS; i += 256)
        atomicMin(&gColMin[(size_t)b * NPTS + jbase + i], ldsColMin[i]);
}

// ---------------------------------------------------------------------------
// Final: deterministic tree-sum of both min arrays, emit max of the two means
// (the int min buffers hold valid non-negative float bit patterns)
// ---------------------------------------------------------------------------
__global__ __launch_bounds__(256) void chamfer_final(
    const float* __restrict__ gColMin,
    const float* __restrict__ gRowMin,
    float* __restrict__ out)
{
    __shared__ float shc[256], shr[256];
    float sc = 0.f, sr = 0.f;
    for (int i = threadIdx.x; i < NB * NPTS; i += 256) {
        sc += gColMin[i];
        sr += gRowMin[i];
    }
    shc[threadIdx.x] = sc;
    shr[threadIdx.x] = sr;
    __syncthreads();
    for (int s2 = 128; s2 > 0; s2 >>= 1) {
        if (threadIdx.x < s2) {
            shc[threadIdx.x] += shc[threadIdx.x + s2];
            shr[threadIdx.x] += shr[threadIdx.x + s2];
        }
        __syncthreads();
    }
    if (threadIdx.x == 0) {
        float inv   = 1.0f / (float)(NB * NPTS);
        float loss1 = shc[0] * inv;   // mean over preds of min-over-gts (col)
        float loss2 = shr[0] * inv;   // mean over gts  of min-over-preds (row)
        out[0] = fmaxf(loss1, loss2);
    }
}

// ---------------------------------------------------------------------------
extern "C" void kernel_launch(void* const* d_in, const int* in_sizes, int n_in,
                              void* d_out, int out_size, void* d_ws, size_t ws_size,
                              hipStream_t stream) {
    (void)in_sizes; (void)n_in; (void)out_size; (void)ws_size;
    const float* preds = (const float*)d_in[0];   // setup_inputs order: preds, gts
    const float* gts   = (const float*)d_in[1];

    int* gColMin = (int*)d_ws;                    // NB*NPTS ints
    int* gRowMin = gColMin + NB * NPTS;           // NB*NPTS ints

    chamfer_init<<<(2 * NB * NPTS + 255) / 256, 256, 0, stream>>>(gColMin);
    chamfer_main<<<dim3(NROWBLOCKS, NB, JSPLIT), 256, 0, stream>>>(
        preds, gts, gColMin, gRowMin);
    chamfer_final<<<1, 256, 0, stream>>>(
        (const float*)gColMin, (const float*)gRowMin, (float*)d_out);
}